// Loss_72481868087527
// MI455X (gfx1250) — compile-verified
//
#include <hip/hip_runtime.h>
#include <cstdint>
#include <math.h>

#define NMODS 6
#define NPRED 30
#define WPB   8                 // waves per block (wave32)
#define BLOCK (WPB * 32)
#define ROWF  (NMODS * NPRED * 2)   // 360 floats = 1440 B per row of reg
#define BUFF  384                   // floats per LDS buffer (1536 B, 96 x 16B chunks)
#define RAD2DEG 57.29577951308232f
#define DEG2RAD 0.017453292519943295f

#if defined(__has_builtin)
#  if __has_builtin(__builtin_amdgcn_global_load_async_to_lds_b128)
#    define HAVE_ASYNC 1
#  endif
#  if __has_builtin(__builtin_amdgcn_s_wait_asynccnt)
#    define HAVE_WAITA 1
#  endif
#endif

typedef int v4i __attribute__((ext_vector_type(4)));
typedef __attribute__((address_space(1))) v4i as1_v4i;
typedef __attribute__((address_space(3))) v4i as3_v4i;

// Stage one full reg row (1440B) into this wave's LDS buffer: 3 x (32 lanes x 16B).
__device__ __forceinline__ void stage_row(const float* __restrict__ reg,
                                          long long rowBase, float* buf, int lane) {
#pragma unroll
  for (int k = 0; k < 3; ++k) {
    const int c  = lane + 32 * k;
    const int cg = (c < 90) ? c : 0;   // clamp: keep EXEC full, junk lands in LDS pad
#if HAVE_ASYNC
    as1_v4i* gp = (as1_v4i*)(uintptr_t)(reg + rowBase + (long long)cg * 4);
    as3_v4i* lp = (as3_v4i*)(uint32_t)(uintptr_t)(buf + c * 4);
    __builtin_amdgcn_global_load_async_to_lds_b128(gp, lp, 0, 0);
#else
    if (c < 90) {
      const float4 v = *(const float4*)(reg + rowBase + (long long)c * 4);
      *(float4*)(buf + c * 4) = v;
    }
#endif
  }
}

__device__ __forceinline__ void waita3() {
#if HAVE_ASYNC
#  if defined(HAVE_WAITA)
  __builtin_amdgcn_s_wait_asynccnt(3);
#  else
  asm volatile("s_wait_asynccnt 3" ::: "memory");
#  endif
#endif
  asm volatile("" ::: "memory");
}

__device__ __forceinline__ void waita0() {
#if HAVE_ASYNC
#  if defined(HAVE_WAITA)
  __builtin_amdgcn_s_wait_asynccnt(0);
#  else
  asm volatile("s_wait_asynccnt 0" ::: "memory");
#  endif
#endif
  asm volatile("" ::: "memory");
}

__global__ __launch_bounds__(BLOCK) void loss_main_kernel(
    const float* __restrict__ reg, const float* __restrict__ cls,
    const float* __restrict__ gt, const unsigned char* __restrict__ has,
    float* __restrict__ acc, int B) {
  __shared__ __align__(16) float smem[WPB * 2 * BUFF];  // 24 KB
  const int lane   = threadIdx.x & 31;
  const int wave   = threadIdx.x >> 5;
  const int gwave  = blockIdx.x * WPB + wave;
  const int nwaves = gridDim.x * WPB;
  float* wbuf = &smem[wave * 2 * BUFF];

  const int  t      = lane;             // timestep owned by this lane
  const bool active = (t < NPRED);
  const int  tc     = active ? t : (NPRED - 1);

  // per-lane accumulators, reduced across the wave once at the end
  float a6x = 0.f, a6y = 0.f, f6x = 0.f, f6y = 0.f;
  float a1x = 0.f, a1y = 0.f, f1x = 0.f, f1y = 0.f;
  float mgnsum = 0.f, regsum = 0.f, nclsf = 0.f, nregf = 0.f;
  const float l0 = (lane == 0) ? 1.f : 0.f;

  int row = gwave;
  int cur = 0;
  if (row < B) stage_row(reg, (long long)row * ROWF, wbuf, lane);

  while (row < B) {
    const int  nxt = row + nwaves;
    const bool hn  = (nxt < B);
    if (hn) {
      stage_row(reg, (long long)nxt * ROWF, wbuf + (cur ^ 1) * BUFF, lane);
      __builtin_prefetch(gt + (long long)nxt * (NPRED * 2), 0, 0);
      __builtin_prefetch(cls + (long long)nxt * NMODS, 0, 0);
      __builtin_prefetch(has + (long long)nxt * NPRED, 0, 0);
    }
    if (hn) waita3(); else waita0();   // current row's 3 async ops complete

    // ---- per-row side data ----
    const float2 g2 = *(const float2*)(gt + (long long)row * (NPRED * 2) + tc * 2);
    const float  gx = g2.x, gy = g2.y;
    const float  hv = active ? (has[(long long)row * NPRED + tc] ? 1.f : 0.f) : 0.f;

    // last_idcs = argmax(has + 0.1*t/30) (first occurrence), valid = max > 1
    float mv = active ? (hv + (0.1f / (float)NPRED) * (float)t) : -1e30f;
    int   mi = t;
#pragma unroll
    for (int off = 16; off > 0; off >>= 1) {
      const float ov = __shfl_xor(mv, off, 32);
      const int   oi = __shfl_xor(mi, off, 32);
      if (ov > mv || (ov == mv && oi < mi)) { mv = ov; mi = oi; }
    }
    const int  last  = mi;
    const bool valid = (mv > 1.0f);

    // heading: seg[t] = atan2(gt[t+1]-gt[t]); seg[t-1] via shuffle
    const float gxn  = __shfl(gx, t + 1, 32);
    const float gyn  = __shfl(gy, t + 1, 32);
    const float segf = atan2f(gyn - gy, gxn - gx);
    const float segb = __shfl(segf, t - 1, 32);
    const float degf = segf * RAD2DEG, degb = segb * RAD2DEG;
    const float hsum = degf + degb;
    const float mid  = ((segf == 0.f) || (segb == 0.f)) ? hsum : 0.5f * hsum;
    float head = (t == 0) ? degf : ((t >= NPRED - 1) ? degb : mid);

    const float g0x = __shfl(gx, 0, 32),         g0y = __shfl(gy, 0, 32);
    const float gLx = __shfl(gx, NPRED - 1, 32), gLy = __shfl(gy, NPRED - 1, 32);
    const float mdx = g0x - gLx, mdy = g0y - gLy;
    const bool moving = sqrtf(mdx * mdx + mdy * mdy) > 2.0f;
    head = moving ? head : 0.f;
    float sth, cth;
    __sincosf(-head * DEG2RAD, &sth, &cth);

    // cls: broadcast 6 scores to all lanes; top1 = first argmax
    const float cvl = (lane < NMODS) ? cls[(long long)row * NMODS + lane] : 0.f;
    float clsv[NMODS];
#pragma unroll
    for (int m = 0; m < NMODS; ++m) clsv[m] = __shfl(cvl, m, 32);
    int top1 = 0; float cbest = clsv[0];
#pragma unroll
    for (int m = 1; m < NMODS; ++m) if (clsv[m] > cbest) { cbest = clsv[m]; top1 = m; }

    const float w  = valid ? hv : 0.f;
    nregf += w;
    const float ga = active ? 1.f : 0.f;
    const float gl = (active && t == NPRED - 1) ? 1.f : 0.f;

    // ---- single streaming pass over this row's reg (from LDS) ----
    const float* rb = wbuf + cur * BUFF;
    float dist2[NMODS], sl1m[NMODS];
#pragma unroll
    for (int m = 0; m < NMODS; ++m) {
      const float2 r2 = *(const float2*)(rb + m * (NPRED * 2) + tc * 2);
      const float dx0 = r2.x - gx, dy0 = r2.y - gy;
      const float d2  = dx0 * dx0 + dy0 * dy0;
      dist2[m] = __shfl(d2, last, 32);               // dist of mode m at last point
      const float adx = fabsf(dx0), ady = fabsf(dy0);
      const float s1 = (adx < 1.f) ? 0.5f * dx0 * dx0 : (adx - 0.5f);
      const float s2 = (ady < 1.f) ? 0.5f * dy0 * dy0 : (ady - 0.5f);
      sl1m[m] = w * (s1 + s2);                       // per-lane smooth-L1 partial
      const float rxr = cth * adx - sth * ady;       // rotated abs-error
      const float ryr = sth * adx + cth * ady;
      const float dex = fabsf(rxr), dey = fabsf(ryr);
      a6x += ga * dex; a6y += ga * dey;
      f6x += gl * dex; f6y += gl * dey;
      const float gm = (m == top1) ? 1.f : 0.f;
      a1x += gm * ga * dex; a1y += gm * ga * dey;
      f1x += gm * gl * dex; f1y += gm * gl * dey;
    }

    // min-dist mode (first occurrence)
    int minm = 0; float md2 = dist2[0];
#pragma unroll
    for (int m = 1; m < NMODS; ++m) if (dist2[m] < md2) { md2 = dist2[m]; minm = m; }
    const float min_dist = sqrtf(md2);

    float srp = sl1m[0], cmin = clsv[0];
#pragma unroll
    for (int m = 1; m < NMODS; ++m) {
      if (m == minm) { srp = sl1m[m]; cmin = clsv[m]; }
    }
#pragma unroll
    for (int off = 16; off > 0; off >>= 1) srp += __shfl_xor(srp, off, 32);

    float nc = 0.f, ms = 0.f;
#pragma unroll
    for (int m = 0; m < NMODS; ++m) {
      const float dm  = sqrtf(dist2[m]);
      const float mgn = cmin - clsv[m];
      const bool mk = (min_dist < 2.0f) && ((dm - min_dist) > 0.2f) && valid && (mgn < 0.2f);
      nc += mk ? 1.f : 0.f;
      ms += mk ? mgn : 0.f;
    }
    nclsf  += l0 * nc;
    mgnsum += l0 * ms;
    regsum += l0 * srp;

    row = nxt; cur ^= 1;
  }

  // fold 12 accumulators across the wave, one atomic each from lane 0
  float vals[12] = {a6x, a6y, f6x, f6y, a1x, a1y, f1x, f1y, mgnsum, regsum, nclsf, nregf};
#pragma unroll
  for (int i = 0; i < 12; ++i) {
    float v = vals[i];
#pragma unroll
    for (int off = 16; off > 0; off >>= 1) v += __shfl_xor(v, off, 32);
    if (lane == 0) atomicAdd(&acc[i], v);
  }
}

__global__ void loss_zero_kernel(float* acc) {
  if (threadIdx.x < 12) acc[threadIdx.x] = 0.f;
}

__global__ void loss_final_kernel(const float* __restrict__ acc,
                                  float* __restrict__ out, int B) {
  if (threadIdx.x != 0) return;
  const float a6x = acc[0], a6y = acc[1], f6x = acc[2], f6y = acc[3];
  const float a1x = acc[4], a1y = acc[5], f1x = acc[6], f1y = acc[7];
  const float mgnsum = acc[8], regsum = acc[9], ncls = acc[10], nreg = acc[11];
  const float cls_loss = 1.0f * (0.2f * ncls - mgnsum);
  const float reg_loss = 1.0f * regsum;
  const float loss = cls_loss / (ncls + 1e-10f) + reg_loss / (nreg + 1e-10f);
  out[0] = loss;  out[1] = cls_loss; out[2] = ncls; out[3] = reg_loss; out[4] = nreg;
  out[5] = a6x;   out[6] = a6y;      out[7] = f6x;  out[8] = f6y;
  out[9]  = (float)((long long)NMODS * B * NPRED);
  out[10] = (float)((long long)NMODS * B);
  out[11] = a1x;  out[12] = a1y;     out[13] = f1x; out[14] = f1y;
  out[15] = (float)((long long)B * NPRED);
  out[16] = (float)B;
}

extern "C" void kernel_launch(void* const* d_in, const int* in_sizes, int n_in,
                              void* d_out, int out_size, void* d_ws, size_t ws_size,
                              hipStream_t stream) {
  const float* reg = (const float*)d_in[0];
  const float* cls = (const float*)d_in[1];
  const float* gt  = (const float*)d_in[2];
  const unsigned char* has = (const unsigned char*)d_in[3];
  const int B = in_sizes[1] / NMODS;   // cls is [B, 6]
  float* acc = (float*)d_ws;
  float* out = (float*)d_out;

  loss_zero_kernel<<<1, 32, 0, stream>>>(acc);
  int blocks = (B + WPB - 1) / WPB;
  if (blocks > 1024) blocks = 1024;
  if (blocks < 1) blocks = 1;
  loss_main_kernel<<<blocks, BLOCK, 0, stream>>>(reg, cls, gt, has, acc, B);
  loss_final_kernel<<<1, 1, 0, stream>>>(acc, out, B);
}